// LocalNCC_metric_49314814492857
// MI455X (gfx1250) — compile-verified
//
#include <hip/hip_runtime.h>
#include <cstddef>

typedef float f4  __attribute__((ext_vector_type(4)));
typedef float v2f __attribute__((ext_vector_type(2)));
typedef float v8f __attribute__((ext_vector_type(8)));

#define B_   8
#define C_   16
#define K_   8
#define H_   512
#define W_   512
#define PLANE_ (H_ * W_)
#define NIMG_  (B_ * C_ * PLANE_)

#ifndef __has_builtin
#define __has_builtin(x) 0
#endif
#if defined(__HIP_DEVICE_COMPILE__) && __has_builtin(__builtin_amdgcn_wmma_f32_16x16x4_f32)
#define HAVE_WMMA 1
#else
#define HAVE_WMMA 0
#endif

// ---------------------------------------------------------------- zero ws
__global__ void k_zero(float* ws) {
    if (threadIdx.x < 16) ws[threadIdx.x] = 0.f;
}

// ------------------------------------------------- per-batch full mask sum
__global__ __launch_bounds__(256)
void k_masksum(const float* __restrict__ mask, float* __restrict__ ws_msum) {
    const int b = blockIdx.y;
    const f4* m4 = (const f4*)(mask + (size_t)b * C_ * PLANE_);
    const int n4 = C_ * PLANE_ / 4;
    float s = 0.f;
    for (int i = blockIdx.x * blockDim.x + threadIdx.x; i < n4;
         i += gridDim.x * blockDim.x) {
        f4 v = m4[i];
        s += v.x + v.y + v.z + v.w;
    }
    __shared__ float red[256];
    red[threadIdx.x] = s;
    __syncthreads();
    for (int off = 128; off > 0; off >>= 1) {
        if ((int)threadIdx.x < off) red[threadIdx.x] += red[threadIdx.x + off];
        __syncthreads();
    }
    if (threadIdx.x == 0) atomicAdd(&ws_msum[b], red[0]);
}

// --------------------------------------------------------- fused NCC tiles
// One wave32 per 16x16 output tile. 4 waves (4 adjacent tiles in x) per block.
// Stage pred/target (18x18 halo) in LDS -> horizontal 3-tap (VALU) into
// 18x16 per quantity -> vertical 3-tap as WMMA f32 16x16x4 band-matrix GEMM:
//   Out(16x16) = Sv(16x20) x H(20x16),  Sv[m][k] = (m <= k <= m+2)
__global__ __launch_bounds__(128)
void k_ncc(const float* __restrict__ img1, const float* __restrict__ img2,
           const float* __restrict__ mask, const int* __restrict__ chan_idx,
           const float* __restrict__ nm, const float* __restrict__ ns,
           float* __restrict__ ws_ncc) {
    __shared__ float s_p[4][18 * 18];
    __shared__ float s_t[4][18 * 18];
    __shared__ float s_H[4][5][18 * 16];

    const int w    = threadIdx.x >> 5;
    const int lane = threadIdx.x & 31;
    const int z    = blockIdx.z;          // b*K_ + k
    const int b    = z >> 3;
    const int k    = z & 7;
    const int y0   = blockIdx.y * 16;
    const int x0   = (blockIdx.x * 4 + w) * 16;

    const int chan  = chan_idx[b * K_ + k];
    const int chan0 = chan_idx[b * K_];
    const float meanv = nm[0];
    const float stdv  = ns[0];

    const size_t base_c  = (size_t)(b * C_ + chan)  * PLANE_;
    const size_t base_m0 = (size_t)(b * C_ + chan0) * PLANE_;

    // ---- stage pred/target with 1-px halo (zero padded at image edge)
    for (int i = lane; i < 18 * 18; i += 32) {
        int yy = i / 18, xx = i - yy * 18;
        int y = y0 - 1 + yy;
        int x = x0 - 1 + xx;
        float p = 0.f, t = 0.f;
        if (y >= 0 && y < H_ && x >= 0 && x < W_) {
            size_t o = (size_t)y * W_ + x;
            float mm = mask[base_c + o] * mask[base_m0 + o];
            float a1 = fmaf(img1[base_c + o], stdv, meanv);
            float a2 = fmaf(img2[base_c + o], stdv, meanv);
            a1 = fminf(fmaxf(a1, 0.f), 1.f);
            a2 = fminf(fmaxf(a2, 0.f), 1.f);
            p = a1 * mm;
            t = a2 * mm;
        }
        s_p[w][i] = p;
        s_t[w][i] = t;
    }
    __syncthreads();

    // ---- horizontal 3-tap on 18 rows x 16 cols, 5 quantities
    for (int i = lane; i < 18 * 16; i += 32) {
        int yy = i >> 4, x = i & 15;
        int c0 = yy * 18 + x;
        float p0 = s_p[w][c0], p1 = s_p[w][c0 + 1], p2 = s_p[w][c0 + 2];
        float t0 = s_t[w][c0], t1 = s_t[w][c0 + 1], t2 = s_t[w][c0 + 2];
        s_H[w][0][i] = t0 + t1 + t2;
        s_H[w][1][i] = p0 + p1 + p2;
        s_H[w][2][i] = t0 * t0 + t1 * t1 + t2 * t2;
        s_H[w][3][i] = p0 * p0 + p1 * p1 + p2 * p2;
        s_H[w][4][i] = t0 * p0 + t1 * p1 + t2 * p2;
    }
    __syncthreads();

    // ---- vertical 3-tap via WMMA band matrix
    const int m  = lane & 15;                 // A: M row / B: N col / D: col
    const int kh = (lane >> 4) << 1;          // K sub-offset per lane half
    v2f afrag[5];
    for (int kk = 0; kk < 5; ++kk) {          // K = 20 (rows 18,19 are A=0 pad)
        int k0 = kk * 4 + kh;
        afrag[kk].x = (k0     >= m && k0     <= m + 2) ? 1.f : 0.f;
        afrag[kk].y = (k0 + 1 >= m && k0 + 1 <= m + 2) ? 1.f : 0.f;
    }

    v8f outq[5];
    for (int q = 0; q < 5; ++q) {
        v8f acc = {};
#if HAVE_WMMA
        for (int kk = 0; kk < 5; ++kk) {
            int k0 = kk * 4 + kh;
            int r0 = (k0     < 18) ? k0       : 17;   // pad rows: A is 0 there
            int r1 = (k0 + 1 < 18) ? (k0 + 1) : 17;
            v2f bf;
            bf.x = s_H[w][q][r0 * 16 + m];
            bf.y = s_H[w][q][r1 * 16 + m];
            acc = __builtin_amdgcn_wmma_f32_16x16x4_f32(
                false, afrag[kk], false, bf, (short)0, acc, false, false);
        }
#else
        for (int v = 0; v < 8; ++v) {         // VALU fallback, same D layout
            int row = v + ((lane >> 4) << 3);
            acc[v] = s_H[w][q][row * 16 + m] + s_H[w][q][(row + 1) * 16 + m] +
                     s_H[w][q][(row + 2) * 16 + m];
        }
#endif
        outq[q] = acc;
    }

    // ---- elementwise NCC on this lane's 8 pixels, then wave reduce
    const float inv9 = 1.0f / 9.0f;
    float lsum = 0.f;
    for (int v = 0; v < 8; ++v) {
        float tS  = outq[0][v], pS  = outq[1][v];
        float t2S = outq[2][v], p2S = outq[3][v], tpS = outq[4][v];
        float cross = tpS - pS * inv9 * tS;
        float tvar  = fmaxf(t2S - tS * inv9 * tS, 0.f);
        float pvar  = fmaxf(p2S - pS * inv9 * pS, 0.f);
        lsum += (cross * cross) / (tvar * pvar + 1e-5f);
    }
    for (int off = 16; off > 0; off >>= 1)
        lsum += __shfl_down(lsum, off, 32);
    if (lane == 0) atomicAdd(&ws_ncc[b], lsum);
}

// -------------------------------------- streaming copy (dst is 4B-aligned)
__global__ __launch_bounds__(256)
void k_copy4(const float* __restrict__ src, float* __restrict__ dst, int n4) {
    const f4* s4 = (const f4*)src;
    const int stride = gridDim.x * blockDim.x;
    for (int i = blockIdx.x * blockDim.x + threadIdx.x; i < n4; i += stride) {
        f4 v = __builtin_nontemporal_load(s4 + i);
        size_t j = (size_t)i * 4;
        __builtin_nontemporal_store(v.x, dst + j);
        __builtin_nontemporal_store(v.y, dst + j + 1);
        __builtin_nontemporal_store(v.z, dst + j + 2);
        __builtin_nontemporal_store(v.w, dst + j + 3);
    }
}

// ------------------------------------------------------------------ metric
__global__ void k_final(const float* __restrict__ ws, float* __restrict__ out) {
    if (threadIdx.x == 0 && blockIdx.x == 0) {
        float s = 0.f;
        for (int b = 0; b < B_; ++b) s += ws[b] / ws[8 + b];
        out[0] = s * (1.0f / B_);
    }
}

extern "C" void kernel_launch(void* const* d_in, const int* in_sizes, int n_in,
                              void* d_out, int out_size, void* d_ws, size_t ws_size,
                              hipStream_t stream) {
    (void)in_sizes; (void)n_in; (void)out_size; (void)ws_size;
    const float* img1     = (const float*)d_in[0];
    const float* mask     = (const float*)d_in[1];
    const float* img2     = (const float*)d_in[2];
    const int*   chan_idx = (const int*)d_in[3];
    const float* nmean    = (const float*)d_in[4];
    const float* nstd     = (const float*)d_in[5];
    // d_in[6] = kernel_size (3 in this workload; kernel hardcodes 3x3)
    float* out = (float*)d_out;
    float* ws  = (float*)d_ws;   // ws[0..7] ncc sums, ws[8..15] mask sums

    k_zero<<<1, 32, 0, stream>>>(ws);
    k_masksum<<<dim3(256, B_), 256, 0, stream>>>(mask, ws + 8);
    k_ncc<<<dim3(W_ / 16 / 4, H_ / 16, B_ * K_), 128, 0, stream>>>(
        img1, img2, mask, chan_idx, nmean, nstd, ws);
    k_copy4<<<4096, 256, 0, stream>>>(img1, out + 1, NIMG_ / 4);
    k_copy4<<<4096, 256, 0, stream>>>(img2, out + 1 + (size_t)NIMG_, NIMG_ / 4);
    k_final<<<1, 32, 0, stream>>>(ws, out);
}